// GraphConstructorGDN2_12206297055833
// MI455X (gfx1250) — compile-verified
//
#include <hip/hip_runtime.h>
#include <stdint.h>
#include <math.h>

// Problem constants (match reference)
#define N_NODES 8192
#define DIM     128
#define TOPK    32

typedef float v2f __attribute__((ext_vector_type(2)));
typedef float v8f __attribute__((ext_vector_type(8)));

// ---------------------------------------------------------------------------
// Kernel 1: per-row inverse L2 norm of w = emb[idx]
// grid = N_NODES blocks, block = DIM(=128) threads
// ---------------------------------------------------------------------------
__global__ void __launch_bounds__(DIM)
norms_kernel(const float* __restrict__ emb, const int* __restrict__ idx,
             float* __restrict__ rinv) {
  __shared__ float s[DIM];
  const int row = blockIdx.x;
  const int t   = threadIdx.x;
  const float* p = emb + (size_t)idx[row] * DIM;
  float x = p[t];
  s[t] = x * x;
  __syncthreads();
  for (int off = DIM / 2; off > 0; off >>= 1) {
    if (t < off) s[t] += s[t + off];
    __syncthreads();
  }
  if (t == 0) rinv[row] = 1.0f / sqrtf(s[0]);
}

// ---------------------------------------------------------------------------
// Kernel 2: cos = (w . w^T) * rinv_i * rinv_j via V_WMMA_F32_16X16X4_F32.
// Block = 256 threads = 8 waves. Each wave owns 4 16x16 C tiles (N=64),
// with their K-loops INTERLEAVED so the 4 accumulators give 4-way WMMA ILP
// (no serial RAW chain through the matrix pipe). Block covers 16x512.
// grid = (N/512, N/16).
//
// Fragment layouts (ISA 7.12.2, 32-bit data):
//   A (16x4):  lanes 0-15: M=lane,   VGPR0=K0, VGPR1=K1
//              lanes16-31: M=lane-16,VGPR0=K2, VGPR1=K3
//   B (4x16):  lanes 0-15: N=lane,   VGPR0=K0, VGPR1=K1
//              lanes16-31: N=lane-16,VGPR0=K2, VGPR1=K3
//   C (16x16): VGPR d: lanes 0-15 -> M=d, lanes 16-31 -> M=d+8; N=lane%16
// ---------------------------------------------------------------------------
#define NSUB 4
__global__ void __launch_bounds__(256)
gemm_kernel(const float* __restrict__ emb, const int* __restrict__ idx,
            const float* __restrict__ rinv, float* __restrict__ out) {
  const int lane  = threadIdx.x & 31;
  const int wave  = threadIdx.x >> 5;
  const int m0    = blockIdx.y * 16;
  const int j0    = blockIdx.x * (8 * NSUB * 16) + wave * (NSUB * 16);
  const int lm    = lane & 15;
  const int khalf = lane >> 4;            // 0: K pair {0,1}; 1: K pair {2,3}

  // A fragments for this wave's 16 rows, all K=128, kept in registers.
  const float* aBase = emb + (size_t)idx[m0 + lm] * DIM + 2 * khalf;
  v2f a[DIM / 4];
  #pragma unroll
  for (int kk = 0; kk < DIM / 4; ++kk)
    a[kk] = *(const v2f*)(aBase + 4 * kk);

  // Row scale factors for this lane's 8 C rows (M = d + 8*khalf).
  float rs[8];
  #pragma unroll
  for (int d = 0; d < 8; ++d)
    rs[d] = rinv[m0 + d + 8 * khalf];

  // Hoist the 4 column base pointers + column scales out of the K-loop.
  const float* bBase[NSUB];
  float cs[NSUB];
  #pragma unroll
  for (int s = 0; s < NSUB; ++s) {
    const int j = j0 + s * 16 + lm;
    bBase[s] = emb + (size_t)idx[j] * DIM + 2 * khalf;
    cs[s] = rinv[j];
  }

  v8f c[NSUB];
  #pragma unroll
  for (int s = 0; s < NSUB; ++s) c[s] = (v8f){};

  // Interleaved K-loop: per kk, 4 independent WMMAs (distinct accumulators)
  // all sharing the same A fragment while it is hot.
  #pragma unroll
  for (int kk = 0; kk < DIM / 4; ++kk) {
    #pragma unroll
    for (int s = 0; s < NSUB; ++s) {
      v2f b = *(const v2f*)(bBase[s] + 4 * kk);
      // 8 args: (neg_a, A, neg_b, B, c_mod, C, reuse_a, reuse_b)
      c[s] = __builtin_amdgcn_wmma_f32_16x16x4_f32(false, a[kk], false, b,
                                                   (short)0, c[s], false, false);
    }
  }

  // Epilogue: scale by rinv_i * rinv_j, store raw cos.
  #pragma unroll
  for (int s = 0; s < NSUB; ++s) {
    const int j = j0 + s * 16 + lm;
    float* o = out + (size_t)(m0 + 8 * khalf) * N_NODES + j;
    #pragma unroll
    for (int d = 0; d < 8; ++d)
      o[(size_t)d * N_NODES] = c[s][d] * rs[d] * cs[s];
  }
}

// ---------------------------------------------------------------------------
// Kernel 3: per-row exact top-32 by |cos| (tie-break: lower column first,
// matching jax.lax.top_k), then masked rewrite of the row.
// One block (256 threads) per row; row staged in 32KB LDS.
// Key = (bits(|v|) << 32) | ~j  -> unique, monotone in (|v|, -j).
// ---------------------------------------------------------------------------
__global__ void __launch_bounds__(256)
topk_mask_kernel(float* __restrict__ out) {
  __shared__ float rowv[N_NODES];                 // 32 KB
  __shared__ unsigned long long red[256];         // 2 KB
  const int row = blockIdx.x;
  const int t   = threadIdx.x;
  float* orow = out + (size_t)row * N_NODES;

  #pragma unroll
  for (int e = 0; e < N_NODES / 256; ++e)
    rowv[e * 256 + t] = orow[e * 256 + t];
  __syncthreads();

  unsigned long long bound = ~0ull;
  for (int r = 0; r < TOPK; ++r) {
    unsigned long long lm = 0ull;
    #pragma unroll
    for (int e = 0; e < N_NODES / 256; ++e) {
      const int j = e * 256 + t;
      const unsigned int ab = __float_as_uint(fabsf(rowv[j]));
      const unsigned long long key =
          ((unsigned long long)ab << 32) | (unsigned int)(0xFFFFFFFFu - j);
      if (key < bound && key > lm) lm = key;
    }
    red[t] = lm;
    __syncthreads();
    for (int s = 128; s > 0; s >>= 1) {
      if (t < s) { if (red[t + s] > red[t]) red[t] = red[t + s]; }
      __syncthreads();
    }
    bound = red[0];      // r-th largest key; after TOPK rounds = 32nd largest
    __syncthreads();
  }

  #pragma unroll
  for (int e = 0; e < N_NODES / 256; ++e) {
    const int j = e * 256 + t;
    const float v = rowv[j];
    const unsigned int ab = __float_as_uint(fabsf(v));
    const unsigned long long key =
        ((unsigned long long)ab << 32) | (unsigned int)(0xFFFFFFFFu - j);
    orow[j] = (key >= bound) ? v : 0.0f;
  }
}

// ---------------------------------------------------------------------------
extern "C" void kernel_launch(void* const* d_in, const int* in_sizes, int n_in,
                              void* d_out, int out_size, void* d_ws, size_t ws_size,
                              hipStream_t stream) {
  const int*   idx = (const int*)d_in[0];    // setup_inputs order: idx, emb
  const float* emb = (const float*)d_in[1];
  float* out  = (float*)d_out;
  float* rinv = (float*)d_ws;                // 8192 floats = 32 KB scratch

  norms_kernel<<<N_NODES, DIM, 0, stream>>>(emb, idx, rinv);

  dim3 ggrid(N_NODES / (8 * NSUB * 16), N_NODES / 16);  // (16, 512)
  gemm_kernel<<<ggrid, 256, 0, stream>>>(emb, idx, rinv, out);

  topk_mask_kernel<<<N_NODES, 256, 0, stream>>>(out);
}